// Text_memory_41162966565358
// MI455X (gfx1250) — compile-verified
//
#include <hip/hip_runtime.h>
#include <hip/hip_bf16.h>

// Problem constants (match reference)
constexpr int NB   = 16;     // batch
constexpr int NC   = 256;    // img channels
constexpr int NH   = 96;     // img H
constexpr int NW   = 96;     // img W
constexpr int FG   = 64;     // feature groups
constexpr int TFC  = 128;    // text feature channels
constexpr float EPSF = 1e-5f;

typedef float v2f __attribute__((ext_vector_type(2)));
typedef float v8f __attribute__((ext_vector_type(8)));

// ---------------------------------------------------------------------------
// Kernel 1: t1raw[b, o*16+ij] = sum_c text[b,c] * w1[c, o, i, j] + b1[o]
// GEMM [16 x 1024 x 128] via V_WMMA_F32_16X16X4_F32. One wave per 16x16 tile.
// Layouts per CDNA5 ISA 7.12.2:
//   A (16x4, 2 VGPR): lanes 0-15 hold {A[m,0],A[m,1]}, lanes 16-31 {A[m,2],A[m,3]}
//   B (4x16, 2 VGPR): lanes 0-15 hold rows {0,1}, lanes 16-31 rows {2,3}, col=lane%16
//   C/D (16x16, 8 VGPR): VGPR r -> row r (lanes 0-15) / row r+8 (lanes 16-31)
// ---------------------------------------------------------------------------
__global__ __launch_bounds__(32) void k_text_gemm(
    const float* __restrict__ text, const float* __restrict__ w1,
    const float* __restrict__ b1, float* __restrict__ t1raw) {
  const int n0   = blockIdx.x * 16;     // output-column tile (0..1023)
  const int lane = threadIdx.x & 31;
  const int half = lane >> 4;           // 0: K{0,1}/rows 0-7, 1: K{2,3}/rows 8-15
  const int mr   = lane & 15;           // A row / B,D column within tile

  v8f acc = {};
  for (int k = 0; k < TFC; k += 4) {
    const int ka = k + 2 * half;
    v2f a, bf;
    a.x  = text[mr * TFC + ka];
    a.y  = text[mr * TFC + ka + 1];
    bf.x = w1[(size_t)ka       * (FG * 16) + n0 + mr];
    bf.y = w1[(size_t)(ka + 1) * (FG * 16) + n0 + mr];
    acc = __builtin_amdgcn_wmma_f32_16x16x4_f32(
        /*neg_a=*/false, a, /*neg_b=*/false, bf,
        /*c_mod=*/(short)0, acc, /*reuse_a=*/false, /*reuse_b=*/false);
  }
  const int   n    = n0 + mr;
  const float bias = b1[n >> 4];        // o = n/16
#pragma unroll
  for (int r = 0; r < 8; ++r) {
    t1raw[(r + 8 * half) * (FG * 16) + n] = acc[r] + bias;
  }
}

// ---------------------------------------------------------------------------
// Kernel 2: BN(train-mode batch stats over 16*16=256 vals per channel)+ReLU
// ---------------------------------------------------------------------------
__global__ __launch_bounds__(256) void k_bn1(
    const float* __restrict__ t1raw, const float* __restrict__ gamma,
    const float* __restrict__ beta, float* __restrict__ t1n) {
  __shared__ float ssum[256];
  __shared__ float ssq[256];
  const int o   = blockIdx.x;           // channel 0..63
  const int tid = threadIdx.x;
  const int b   = tid >> 4;
  const int ij  = tid & 15;
  const float x = t1raw[b * (FG * 16) + o * 16 + ij];
  ssum[tid] = x;
  ssq[tid]  = x * x;
  __syncthreads();
  for (int s = 128; s > 0; s >>= 1) {
    if (tid < s) { ssum[tid] += ssum[tid + s]; ssq[tid] += ssq[tid + s]; }
    __syncthreads();
  }
  const float mean  = ssum[0] * (1.0f / 256.0f);
  const float var   = ssq[0] * (1.0f / 256.0f) - mean * mean;
  const float scale = rsqrtf(var + EPSF) * gamma[o];
  const float y     = (x - mean) * scale + beta[o];
  t1n[b * (FG * 16) + o * 16 + ij] = fmaxf(y, 0.0f);
}

// ---------------------------------------------------------------------------
// Kernel 3: ConvTranspose2d(FG,FG,4,stride=2,pad=1): [16,64,4,4]->[16,64,8,8]
// out[b,o,y,x] = b2[o] + sum_i sum_{yi,xi} t1n[b,i,yi,xi]*w2[i,o,kh,kw],
//   kh = y - 2*yi + 1 in [0,3], kw = x - 2*xi + 1 in [0,3]
// ---------------------------------------------------------------------------
__global__ __launch_bounds__(256) void k_convt2(
    const float* __restrict__ t1n, const float* __restrict__ w2,
    const float* __restrict__ b2, float* __restrict__ t2raw) {
  const int gid = blockIdx.x * 256 + threadIdx.x;  // < 65536
  const int b = gid >> 12;
  const int o = (gid >> 6) & 63;
  const int y = (gid >> 3) & 7;
  const int x = gid & 7;

  int yis[2], khs[2], nyi = 0;
  for (int yi = 0; yi < 4; ++yi) {
    const int kh = y - 2 * yi + 1;
    if (kh >= 0 && kh < 4) { yis[nyi] = yi; khs[nyi] = kh; ++nyi; }
  }
  int xis[2], kws[2], nxi = 0;
  for (int xi = 0; xi < 4; ++xi) {
    const int kw = x - 2 * xi + 1;
    if (kw >= 0 && kw < 4) { xis[nxi] = xi; kws[nxi] = kw; ++nxi; }
  }

  float acc = b2[o];
  for (int i = 0; i < FG; ++i) {
    const float* tin  = t1n + b * (FG * 16) + i * 16;
    const float* wrow = w2 + (size_t)i * (FG * 16) + o * 16;  // w2:(in,out,4,4)
    for (int a = 0; a < nyi; ++a)
      for (int c2 = 0; c2 < nxi; ++c2)
        acc = fmaf(tin[yis[a] * 4 + xis[c2]], wrow[khs[a] * 4 + kws[c2]], acc);
  }
  t2raw[gid] = acc;  // layout [b][o][y*8+x]
}

// ---------------------------------------------------------------------------
// Kernel 4: BN over 16*64=1024 vals per channel + ReLU -> templates t2n[b][o][64]
// ---------------------------------------------------------------------------
__global__ __launch_bounds__(256) void k_bn2(
    const float* __restrict__ t2raw, const float* __restrict__ gamma,
    const float* __restrict__ beta, float* __restrict__ t2n) {
  __shared__ float ssum[256];
  __shared__ float ssq[256];
  const int o   = blockIdx.x;
  const int tid = threadIdx.x;
  float xs[4];
  float s = 0.0f, q = 0.0f;
#pragma unroll
  for (int j = 0; j < 4; ++j) {
    const int f = tid + j * 256;        // 0..1023
    const int b = f >> 6, p = f & 63;
    const float x = t2raw[b * (FG * 64) + o * 64 + p];
    xs[j] = x; s += x; q += x * x;
  }
  ssum[tid] = s;
  ssq[tid]  = q;
  __syncthreads();
  for (int st = 128; st > 0; st >>= 1) {
    if (tid < st) { ssum[tid] += ssum[tid + st]; ssq[tid] += ssq[tid + st]; }
    __syncthreads();
  }
  const float mean  = ssum[0] * (1.0f / 1024.0f);
  const float var   = ssq[0] * (1.0f / 1024.0f) - mean * mean;
  const float scale = rsqrtf(var + EPSF) * gamma[o];
  const float bb    = beta[o];
#pragma unroll
  for (int j = 0; j < 4; ++j) {
    const int f = tid + j * 256;
    const int b = f >> 6, p = f & 63;
    t2n[b * (FG * 64) + o * 64 + p] = fmaxf((xs[j] - mean) * scale + bb, 0.0f);
  }
}

// ---------------------------------------------------------------------------
// Kernel 5 (the 151 MB streaming kernel): per (b,c) plane, depthwise 8x8
// cross-correlation (stride 4, valid -> 23x23) against template fg=c/4,
// global spatial max, relu, /64. One workgroup per plane; plane staged in LDS.
// ---------------------------------------------------------------------------
__global__ __launch_bounds__(256) void k_corr_max(
    const float* __restrict__ img, const float* __restrict__ t2n,
    float* __restrict__ out) {
  __shared__ float plane[NH * NW];      // 9216 floats = 36 KB
  __shared__ float kern[64];
  __shared__ float red[256];

  const int bid = blockIdx.x;           // 0..4095
  const int b   = bid >> 8;
  const int c   = bid & 255;
  const int fg  = c >> 2;               // rep = NC/FG = 4
  const int tid = threadIdx.x;

  if (tid < 64) kern[tid] = t2n[((size_t)b * FG + fg) * 64 + tid];

  // Coalesced float4 staging: 9216 floats = 2304 float4 = 256 threads * 9
  const float4* src = (const float4*)(img + (size_t)bid * (NH * NW));
  float4* dst = (float4*)plane;
#pragma unroll
  for (int i = 0; i < 9; ++i) dst[tid + i * 256] = src[tid + i * 256];
  __syncthreads();

  float m = -3.402823466e+38f;
  for (int o = tid; o < 23 * 23; o += 256) {
    const int oy = o / 23, ox = o - oy * 23;
    const float* p = plane + (oy * 4) * NW + ox * 4;
    float acc = 0.0f;
#pragma unroll
    for (int ky = 0; ky < 8; ++ky)
#pragma unroll
      for (int kx = 0; kx < 8; ++kx)
        acc = fmaf(p[ky * NW + kx], kern[ky * 8 + kx], acc);
    m = fmaxf(m, acc);
  }
  red[tid] = m;
  __syncthreads();
  for (int s = 128; s > 0; s >>= 1) {
    if (tid < s) red[tid] = fmaxf(red[tid], red[tid + s]);
    __syncthreads();
  }
  if (tid == 0) out[bid] = fmaxf(red[0], 0.0f) * (1.0f / 64.0f);
}

// ---------------------------------------------------------------------------
extern "C" void kernel_launch(void* const* d_in, const int* in_sizes, int n_in,
                              void* d_out, int out_size, void* d_ws, size_t ws_size,
                              hipStream_t stream) {
  (void)in_sizes; (void)n_in; (void)out_size; (void)ws_size;
  const float* img  = (const float*)d_in[0];
  const float* text = (const float*)d_in[1];
  const float* w1   = (const float*)d_in[2];
  const float* b1   = (const float*)d_in[3];
  const float* g1   = (const float*)d_in[4];
  const float* be1  = (const float*)d_in[5];
  const float* w2   = (const float*)d_in[6];
  const float* b2   = (const float*)d_in[7];
  const float* g2   = (const float*)d_in[8];
  const float* be2  = (const float*)d_in[9];
  float* out = (float*)d_out;

  float* ws    = (float*)d_ws;
  float* t1raw = ws;                     // 16*1024
  float* t1n   = ws + 16384;             // 16*1024
  float* t2raw = ws + 32768;             // 16*64*64
  float* t2n   = ws + 98304;             // 16*64*64  (total 640 KB)

  k_text_gemm<<<64,   32,  0, stream>>>(text, w1, b1, t1raw);
  k_bn1      <<<64,   256, 0, stream>>>(t1raw, g1, be1, t1n);
  k_convt2   <<<256,  256, 0, stream>>>(t1n, w2, b2, t2raw);
  k_bn2      <<<64,   256, 0, stream>>>(t2raw, g2, be2, t2n);
  k_corr_max <<<4096, 256, 0, stream>>>(img, t2n, out);
}